// FixedRadiusNearNeighbors_7301444403892
// MI455X (gfx1250) — compile-verified
//
#include <hip/hip_runtime.h>
#include <stdint.h>

// Fixed-radius near neighbors via V_WMMA_F32_16X16X4_F32 (gfx1250).
//   sqrdist(i,j) = sq_i + sq_j - 2*dot(p_i,p_j)
// A row (16x4)  = [-2x_i, -2y_i, -2z_i, 1]
// B col (4x16)  = [ x_j ,  y_j ,  z_j , sq_j]
// D = A*B       = sq_j - 2*dot   ; compare  D <= R^2 - sq_i  (per-row threshold)
//
// B fragments are staged in LDS (320KB/WGP on CDNA5), pre-swizzled into the
// WMMA 4x16 f32 lane layout; the hot loop double-buffers the ds_load_b64 so
// LDS latency hides behind the wmma/compare/route work.

typedef __attribute__((ext_vector_type(2))) float v2f;
typedef __attribute__((ext_vector_type(8))) float v8f;

#define NPTS        8192
#define NNEI        32
#define RADIUS2     0.01f
#define CHUNK       2048          // points staged per LDS round (32 KB)
#define CHUNK_TILES (CHUNK / 16)  // 128 WMMA column tiles per round

static __device__ __forceinline__ uint32_t ballot32(bool p) {
#if __has_builtin(__builtin_amdgcn_ballot_w32)
  return __builtin_amdgcn_ballot_w32(p);
#else
  return (uint32_t)__ballot(p);
#endif
}

__global__ __launch_bounds__(256) void frnn_wmma_kernel(const float* __restrict__ pos,
                                                        int* __restrict__ out) {
  // Per tile T: 64 floats = [x,y interleaved for n=0..15 | z,sq interleaved]
  __shared__ float smem[CHUNK * 4];  // 32 KB

  const int tid   = threadIdx.x;
  const int lane  = tid & 31;
  const int wave  = blockIdx.x * (blockDim.x >> 5) + (tid >> 5);
  const int rowBase = wave * 32;              // flat (b*N + i) of first owned row
  const int b     = rowBase >> 13;            // / 8192 (blocks are batch-aligned)
  const int iBase = rowBase & (NPTS - 1);
  const float* __restrict__ posB = pos + (size_t)b * NPTS * 3;

  const int  mm = lane & 15;
  const bool hi = (lane & 16) != 0;

  // ---------- A fragments: two 16x4 f32 tiles (ISA layout: V0=K0/K2, V1=K1/K3) ----------
  const float x0 = posB[(iBase + mm) * 3 + 0];
  const float y0 = posB[(iBase + mm) * 3 + 1];
  const float z0 = posB[(iBase + mm) * 3 + 2];
  const float x1 = posB[(iBase + 16 + mm) * 3 + 0];
  const float y1 = posB[(iBase + 16 + mm) * 3 + 1];
  const float z1 = posB[(iBase + 16 + mm) * 3 + 2];

  v2f a0, a1;
  a0.x = hi ? (-2.0f * z0) : (-2.0f * x0);
  a0.y = hi ? 1.0f         : (-2.0f * y0);
  a1.x = hi ? (-2.0f * z1) : (-2.0f * x1);
  a1.y = hi ? 1.0f         : (-2.0f * y1);

  // ---------- per-VGPR row thresholds: thr[r] = R^2 - sq(row r + 8*hi) ----------
  const float sqI0 = x0 * x0 + y0 * y0 + z0 * z0;
  const float sqI1 = x1 * x1 + y1 * y1 + z1 * z1;
  const int srcBase = (lane & 16) >> 1;        // 0 for lanes 0-15, 8 for lanes 16-31
  float thr0[8], thr1[8];
#pragma unroll
  for (int r = 0; r < 8; ++r) {
    thr0[r] = RADIUS2 - __shfl(sqI0, srcBase + r, 32);
    thr1[r] = RADIUS2 - __shfl(sqI1, srcBase + r, 32);
  }

  // ---------- per-lane owned output row ----------
  int* __restrict__ outRow = out + (size_t)(rowBase + lane) * NNEI;
  int  cnt   = 0;
  int  first = iBase + lane;                   // self is always within radius

  const v8f czero = {};
  bool waveDone = false;

  for (int cb = 0; cb < NPTS; cb += CHUNK) {
    __syncthreads();
    // ---- cooperative fill: swizzle CHUNK points into WMMA B-fragment layout ----
#pragma unroll
    for (int p = tid; p < CHUNK; p += 256) {
      const float* __restrict__ pj = posB + (size_t)(cb + p) * 3;
      const float xj = pj[0];
      const float yj = pj[1];
      const float zj = pj[2];
      __builtin_prefetch(posB + (size_t)(cb + p + CHUNK) * 3, 0, 0);
      const float sqj = xj * xj + yj * yj + zj * zj;
      const int T = p >> 4, n = p & 15;
      v2f xy = {xj, yj};
      v2f zs = {zj, sqj};
      *(v2f*)&smem[T * 64 + n * 2]      = xy;
      *(v2f*)&smem[T * 64 + 32 + n * 2] = zs;
    }
    __syncthreads();

    if (waveDone) continue;                    // wave-uniform; still hits barriers

    const float* __restrict__ sbase = smem + lane * 2;
    // software-pipelined B fragment: one ds_load_b64 in flight ahead of use
    v2f bf = *(const v2f*)(sbase);
#pragma unroll 2
    for (int T = 0; T < CHUNK_TILES; ++T) {
      // prefetch next tile (index wrapped -> always-valid, branch-free address)
      const v2f bfn = *(const v2f*)(sbase + ((T + 1) & (CHUNK_TILES - 1)) * 64);

      v8f d0 = __builtin_amdgcn_wmma_f32_16x16x4_f32(false, a0, false, bf,
                                                     (short)0, czero, false, false);
      v8f d1 = __builtin_amdgcn_wmma_f32_16x16x4_f32(false, a1, false, bf,
                                                     (short)0, czero, false, false);

      // ballot each result VGPR: bits 0-15 -> row r cols, bits 16-31 -> row r+8 cols
      uint32_t bal0[8], bal1[8];
#pragma unroll
      for (int r = 0; r < 8; ++r) {
        bal0[r] = ballot32(d0[r] <= thr0[r]);
        bal1[r] = ballot32(d1[r] <= thr1[r]);
      }

      // route the 16-bit column mask to its owner lane (lane owns row rowBase+lane)
      const int r = mm & 7;
      uint32_t s0 = bal0[0], s1 = bal1[0];
#pragma unroll
      for (int k = 1; k < 8; ++k) {
        s0 = (r == k) ? bal0[k] : s0;
        s1 = (r == k) ? bal1[k] : s1;
      }
      const uint32_t bsel = hi ? s1 : s0;
      uint32_t mask16 = ((mm & 8) ? (bsel >> 16) : bsel) & 0xFFFFu;

      // append ascending column indices until this row has 32 neighbors
      const int jb = cb + T * 16;
      while (mask16 && cnt < NNEI) {
        const int n = __builtin_ctz(mask16);
        mask16 &= mask16 - 1;
        const int idx = jb + n;
        if (cnt == 0) first = idx;
        outRow[cnt] = idx;
        ++cnt;
      }

      bf = bfn;
    }

    // per-chunk wave-uniform early exit flag
    waveDone = (ballot32(cnt >= NNEI) == 0xFFFFFFFFu);
  }

  // pad remaining slots with first neighbor (reference semantics)
  for (int k = cnt; k < NNEI; ++k) outRow[k] = first;
}

extern "C" void kernel_launch(void* const* d_in, const int* in_sizes, int n_in,
                              void* d_out, int out_size, void* d_ws, size_t ws_size,
                              hipStream_t stream) {
  const float* pos = (const float*)d_in[0];   // (B, N, 3) float32
  // d_in[1] (centroids) is unused by the reference computation.
  int* out = (int*)d_out;                     // (B, N, 32) int32

  const int totalRows = in_sizes[0] / 3;      // B * N = 32768
  const int waves     = totalRows / 32;       // 32 rows per wave
  const int blocks    = waves / 8;            // 8 waves (256 threads) per block

  frnn_wmma_kernel<<<blocks, 256, 0, stream>>>(pos, out);
}